// RelationGAT_13331578487506
// MI455X (gfx1250) — compile-verified
//
#include <hip/hip_runtime.h>

// CDNA5 / gfx1250: wave32, WMMA 16x16 tiles.
typedef float v2f __attribute__((ext_vector_type(2)));
typedef float v8f __attribute__((ext_vector_type(8)));

#define NB     1024   // batch (rows of A/D, sessions)
#define DIM    100    // embedding dim
#define NPAD   112    // DIM padded to 7 * 16
#define LSEQ   200    // items per session
#define KD     1024   // inner GEMM dimension

// ---------------------------------------------------------------------------
// Kernel 1: rel[b, d] = (sum_l emb[items[b,l]][d]) / session_len[b]
// emb has an implicit zero row at index 0; item_embedding row i backs index i+1.
// Cols 100..111 are written as exact zeros (padding for WMMA N-tiles).
// ---------------------------------------------------------------------------
__global__ void gather_pool_kernel(const float* __restrict__ item_emb,
                                   const long long* __restrict__ items,
                                   const float* __restrict__ session_len,
                                   float* __restrict__ rel) {
    __shared__ long long sidx[LSEQ];
    const int b = blockIdx.x;
    for (int l = threadIdx.x; l < LSEQ; l += blockDim.x)
        sidx[l] = items[(size_t)b * LSEQ + l];
    __syncthreads();

    const int d = threadIdx.x;
    if (d >= NPAD) return;
    float acc = 0.0f;
    if (d < DIM) {
        #pragma unroll 4
        for (int l = 0; l < LSEQ; ++l) {
            const long long idx = sidx[l];
            if (idx > 0)  // index 0 -> zero padding row
                acc += item_emb[(size_t)(idx - 1) * DIM + d];
        }
        acc /= session_len[b];
    }
    rel[b * NPAD + d] = acc;
}

// ---------------------------------------------------------------------------
// Kernel 2: tmp = A @ rel   ([1024,1024] x [1024,112] -> [1024,112], fp32)
// One wave32 per 16x16 output tile, V_WMMA_F32_16X16X4_F32, K-step 4.
// A-frag per ISA layout: lane<16 holds K={0,1}, lane>=16 holds K={2,3} of its
// row -> one float2 (global_load_b64) per lane per step. B-frag mirrors it.
// ---------------------------------------------------------------------------
__global__ void gemm_A_rel_kernel(const float* __restrict__ A,
                                  const float* __restrict__ rel,
                                  float* __restrict__ tmp) {
    const int wave = threadIdx.x >> 5;      // 4 waves / block
    const int lane = threadIdx.x & 31;
    const int tile = blockIdx.x * 4 + wave; // 0..447 = 64 Mtiles * 7 Ntiles
    const int tm = tile / 7;
    const int tn = tile % 7;
    const int half = lane >> 4;             // K sub-pair selector
    const int l16  = lane & 15;

    const float* arow = A   + (size_t)(tm * 16 + l16) * KD + 2 * half;
    const float* bcol = rel + (size_t)(2 * half) * NPAD + tn * 16 + l16;

    v8f c = {0.f, 0.f, 0.f, 0.f, 0.f, 0.f, 0.f, 0.f};
    #pragma unroll 4
    for (int k = 0; k < KD; k += 4) {
        v2f a = *(const v2f*)(arow + k);            // A[row, k+2h .. k+2h+1]
        v2f b;
        b.x = bcol[(size_t)k * NPAD];               // rel[k+2h,   col]
        b.y = bcol[(size_t)(k + 1) * NPAD];         // rel[k+2h+1, col]
        c = __builtin_amdgcn_wmma_f32_16x16x4_f32(false, a, false, b,
                                                  (short)0, c, false, false);
    }

    const int col = tn * 16 + l16;
    #pragma unroll
    for (int i = 0; i < 8; ++i) {
        const int row = tm * 16 + i + 8 * half;     // C/D layout: v_i -> M=i / i+8
        tmp[(size_t)row * NPAD + col] = c[i];
    }
}

// ---------------------------------------------------------------------------
// Kernel 3: out = l2norm_rows(selu(D @ tmp))[:, :100]
// One block = one 16-row M-tile; 7 waves cover the 7 N-tiles so the whole row
// lives in the block. Row sum-of-squares via LDS ds_add_f32, then rsqrt.
// Padding cols are exact zeros -> selu(0)=0 -> contribute 0 to the norm.
// ---------------------------------------------------------------------------
__global__ void gemm_D_selu_norm_kernel(const float* __restrict__ Dm,
                                        const float* __restrict__ tmp,
                                        float* __restrict__ out) {
    __shared__ float normsq[16];
    const int wave = threadIdx.x >> 5;  // 0..6 = N-tile
    const int lane = threadIdx.x & 31;
    const int tm = blockIdx.x;
    const int tn = wave;
    const int half = lane >> 4;
    const int l16  = lane & 15;

    if (threadIdx.x < 16) normsq[threadIdx.x] = 0.0f;
    __syncthreads();

    const float* arow = Dm  + (size_t)(tm * 16 + l16) * KD + 2 * half;
    const float* bcol = tmp + (size_t)(2 * half) * NPAD + tn * 16 + l16;

    v8f c = {0.f, 0.f, 0.f, 0.f, 0.f, 0.f, 0.f, 0.f};
    #pragma unroll 4
    for (int k = 0; k < KD; k += 4) {
        v2f a = *(const v2f*)(arow + k);
        v2f b;
        b.x = bcol[(size_t)k * NPAD];
        b.y = bcol[(size_t)(k + 1) * NPAD];
        c = __builtin_amdgcn_wmma_f32_16x16x4_f32(false, a, false, b,
                                                  (short)0, c, false, false);
    }

    const float kScale = 1.0507009873554805f;
    const float kAlpha = 1.6732632423543772f;
    float s[8];
    #pragma unroll
    for (int i = 0; i < 8; ++i) {
        const float x = c[i];
        const float v = kScale * ((x > 0.0f) ? x : kAlpha * (__expf(x) - 1.0f));
        s[i] = v;
        atomicAdd(&normsq[i + 8 * half], v * v);    // ds_add_f32
    }
    __syncthreads();

    const int col = tn * 16 + l16;
    #pragma unroll
    for (int i = 0; i < 8; ++i) {
        const int r = i + 8 * half;
        const float rn = rsqrtf(normsq[r]);
        if (col < DIM)
            out[(size_t)(tm * 16 + r) * DIM + col] = s[i] * rn;
    }
}

// ---------------------------------------------------------------------------
extern "C" void kernel_launch(void* const* d_in, const int* in_sizes, int n_in,
                              void* d_out, int out_size, void* d_ws, size_t ws_size,
                              hipStream_t stream) {
    const float*     item_emb = (const float*)d_in[0];     // [50000,100] f32
    const long long* items    = (const long long*)d_in[1]; // [1024,200] int64
    const float*     A        = (const float*)d_in[2];     // [1024,1024] f32
    const float*     Dm       = (const float*)d_in[3];     // [1024,1024] f32
    const float*     slen     = (const float*)d_in[4];     // [1024,1]   f32
    // d_in[5] = target_embedding: unused by the reference output.
    float* out = (float*)d_out;                            // [1024,100] f32

    float* rel = (float*)d_ws;            // [1024,112] f32  (448 KB)
    float* tmp = rel + NB * NPAD;         // [1024,112] f32  (448 KB)

    // 1) gather + mean-pool (pads cols 100..111 with zeros)
    gather_pool_kernel<<<NB, 128, 0, stream>>>(item_emb, items, slen, rel);

    // 2) tmp = A @ rel     (associativity rewrite: D@(A@rel) == (D@A)@rel, 5.6x fewer FLOPs)
    gemm_A_rel_kernel<<<112, 128, 0, stream>>>(A, rel, tmp);   // 448 tiles / 4 waves

    // 3) out = rownorm(selu(D @ tmp))[:, :100]
    gemm_D_selu_norm_kernel<<<64, 224, 0, stream>>>(Dm, tmp, out); // 64 M-tiles x 7 waves
}